// GetDensity_37881611551298
// MI455X (gfx1250) — compile-verified
//
#include <hip/hip_runtime.h>

typedef __attribute__((ext_vector_type(2))) float v2f;
typedef __attribute__((ext_vector_type(8))) float v8f;

#define NW    8          // nwave
#define TP    13         // totpara = 1 + 3 + 9
#define NORB  128        // norbit
#define AEL   (TP * NW)  // 104 floats per atom in sumw

// ---------------------------------------------------------------------------
// Kernel 1: zero the sumw scatter buffer (d_ws is not re-poisoned per call,
// and kernel_launch must be deterministic, so zero it every launch).
// ---------------------------------------------------------------------------
__global__ void zero_ws_kernel(float* __restrict__ ws, int n) {
    int i = blockIdx.x * blockDim.x + threadIdx.x;
    if (i < n) ws[i] = 0.0f;
}

// ---------------------------------------------------------------------------
// Kernel 2: per-edge radial/angular terms, scatter-add into sumw[N,13,8]
// via f32 global atomics (sumw = 1.7MB, L2-resident on the 192MB L2).
// ---------------------------------------------------------------------------
__global__ __launch_bounds__(256) void edge_kernel(
    const float* __restrict__ cart,     // [N,3]
    const int*   __restrict__ species,  // [N]
    const int*   __restrict__ aidx,     // [2,E] flat
    const float* __restrict__ shifts,   // [E,3]
    const float* __restrict__ rs,       // [S,8]
    const float* __restrict__ inta,     // [S,8]
    const float* __restrict__ params,   // [S,8]
    float* __restrict__ sumw,           // [N,13,8]
    int E, int P, int A)
{
    int e = blockIdx.x * blockDim.x + threadIdx.x;
    if (e >= E) return;

    int b    = e / P;
    int base = b * A;
    int i0   = aidx[e]     + base;   // center atom (scatter target)
    int i1   = aidx[E + e] + base;   // neighbor atom

    float sx = shifts[3 * e + 0];
    float sy = shifts[3 * e + 1];
    float sz = shifts[3 * e + 2];
    float valid = (sx > -1e10f && sy > -1e10f && sz > -1e10f) ? 1.0f : 0.0f;

    float dx = cart[3 * i0 + 0] - cart[3 * i1 + 0] + sx;
    float dy = cart[3 * i0 + 1] - cart[3 * i1 + 1] + sy;
    float dz = cart[3 * i0 + 2] - cart[3 * i1 + 2] + sz;
    float dist = sqrtf(dx * dx + dy * dy + dz * dz);

    // cutoff-cosine squared
    float cc = 0.5f * cosf(dist * 0.6283185307179586f) + 0.5f;  // pi/CUTOFF = pi/5
    float fc = cc * cc;

    int sp = species[i1];
    float cw = (sp >= 0) ? 1.0f : 0.0f;
    int  spc = (sp >= 0) ? sp : 0;

    // g[k] = exp(inta*(d-rs)^2) * params * valid * (species>=0)
    float g[NW];
    #pragma unroll
    for (int k = 0; k < NW; ++k) {
        float t = dist - rs[spc * NW + k];
        g[k] = __expf(inta[spc * NW + k] * t * t) * params[spc * NW + k] * valid * cw;
    }

    // angular: [fc] ++ fc*dv[p] ++ fc*dv[p]*dv[q]
    float dv[3] = {dx, dy, dz};
    float ang[TP];
    ang[0] = fc;
    #pragma unroll
    for (int p = 0; p < 3; ++p) ang[1 + p] = fc * dv[p];
    #pragma unroll
    for (int p = 0; p < 3; ++p)
        #pragma unroll
        for (int q = 0; q < 3; ++q)
            ang[4 + p * 3 + q] = ang[1 + p] * dv[q];

    float* dst = sumw + (long)i0 * AEL;
    #pragma unroll
    for (int j = 0; j < TP; ++j) {
        float aj = ang[j];
        #pragma unroll
        for (int k = 0; k < NW; ++k)
            atomicAdd(dst + j * NW + k, aj * g[k]);
    }
}

// ---------------------------------------------------------------------------
// Kernel 3: density[n,m] = sum_j ( sum_k sumw[n,j,k] * hyper[ip(j),k,m] )^2
// One wave32 computes a 16x16 (atoms x orbits) tile with V_WMMA_F32_16X16X4_F32:
// K=8 is two chained K=4 WMMAs (full f32, exact vs the f32 reference).
// B depends only on ip(j) in {0,1,2}, so B fragments are hoisted per group.
// Block = 256 threads = 8 waves = all 8 orbit tiles of one atom tile.
// ---------------------------------------------------------------------------
__global__ __launch_bounds__(256) void density_wmma_kernel(
    const float* __restrict__ sumw,    // [N,13,8]
    const float* __restrict__ hyper,   // [3,8,128]
    float* __restrict__ out,           // [N,128]
    int Natoms)
{
    int tile = blockIdx.x;              // 16-atom tile
    int wave = threadIdx.x >> 5;        // orbit tile 0..7
    int lane = threadIdx.x & 31;

    int mrow  = tile * 16 + (lane & 15);      // A-fragment row (M)
    int khalf = (lane < 16) ? 0 : 2;          // K pair held by this lane half
    int coln  = wave * 16 + (lane & 15);      // B-fragment column (N)

    v8f dacc = {0.f, 0.f, 0.f, 0.f, 0.f, 0.f, 0.f, 0.f};

    const int jstart[3] = {0, 1, 4};
    const int jend[3]   = {1, 4, 13};

    for (int ip = 0; ip < 3; ++ip) {
        // B fragments for this hyper slab: b0 covers K=0..3, b1 covers K=4..7
        const float* hb = hyper + ip * NW * NORB;
        v2f b0, b1;
        b0.x = hb[(khalf + 0) * NORB + coln];
        b0.y = hb[(khalf + 1) * NORB + coln];
        b1.x = hb[(4 + khalf + 0) * NORB + coln];
        b1.y = hb[(4 + khalf + 1) * NORB + coln];

        for (int j = jstart[ip]; j < jend[ip]; ++j) {
            const float* sa = sumw + (long)mrow * AEL + j * NW;
            v2f a0, a1;
            a0.x = sa[khalf + 0];
            a0.y = sa[khalf + 1];
            a1.x = sa[4 + khalf + 0];
            a1.y = sa[4 + khalf + 1];

            v8f c = {0.f, 0.f, 0.f, 0.f, 0.f, 0.f, 0.f, 0.f};
            // D = A(16x4) x B(4x16) + C, chained over the two K halves
            c = __builtin_amdgcn_wmma_f32_16x16x4_f32(
                    false, a0, false, b0, (short)0, c, false, false);
            c = __builtin_amdgcn_wmma_f32_16x16x4_f32(
                    false, a1, false, b1, (short)0, c, false, false);
            dacc += c * c;  // elementwise square-accumulate over j
        }
    }

    // C/D layout: VGPR v -> row (v or v+8), lane -> column
    int rbase = tile * 16 + ((lane < 16) ? 0 : 8);
    #pragma unroll
    for (int v = 0; v < 8; ++v)
        out[(rbase + v) * NORB + coln] = dacc[v];
}

// ---------------------------------------------------------------------------
extern "C" void kernel_launch(void* const* d_in, const int* in_sizes, int n_in,
                              void* d_out, int out_size, void* d_ws, size_t ws_size,
                              hipStream_t stream) {
    const float* cart    = (const float*)d_in[0];
    // d_in[1] = num_atoms (unused by the reference math)
    const int*   species = (const int*)  d_in[2];
    const int*   aidx    = (const int*)  d_in[3];
    const float* shifts  = (const float*)d_in[4];
    const float* rs      = (const float*)d_in[5];
    const float* inta    = (const float*)d_in[6];
    const float* params  = (const float*)d_in[7];
    const float* hyper   = (const float*)d_in[8];

    int B = in_sizes[1];          // 32
    int N = in_sizes[2];          // B*A = 4096
    int A = N / B;                // 128
    int E = in_sizes[3] / 2;      // B*P = 262144
    int P = E / B;                // 8192

    float* sumw = (float*)d_ws;   // [N,13,8] f32 = 1.7 MB
    int swn = N * AEL;

    zero_ws_kernel<<<(swn + 255) / 256, 256, 0, stream>>>(sumw, swn);

    edge_kernel<<<(E + 255) / 256, 256, 0, stream>>>(
        cart, species, aidx, shifts, rs, inta, params, sumw, E, P, A);

    density_wmma_kernel<<<N / 16, 256, 0, stream>>>(
        sumw, hyper, (float*)d_out, N);
}